// Free_Block_clip_54803782697343
// MI455X (gfx1250) — compile-verified
//
#include <hip/hip_runtime.h>

// Problem constants (match reference)
constexpr int B_ = 32, C_ = 8, D_ = 512, T_ = 256, O_ = 128, E_ = 16;
// LDS padding to avoid 16-way bank conflicts on A-fragment row reads
constexpr int XM_STRIDE = D_ + 4;   // 516: row stride of xm tile in LDS
constexpr int TO_STRIDE = O_ + 4;   // 132: row stride of tokens/h tiles in LDS

typedef __attribute__((ext_vector_type(2))) float v2f;
typedef __attribute__((ext_vector_type(4))) float v4f;
typedef __attribute__((ext_vector_type(8))) float v8f;

static __device__ __forceinline__ v8f wmma_f32_16x16x4(v2f a, v2f b, v8f c) {
  // D = A(16x4 f32) * B(4x16 f32) + C(16x16 f32); full fp32 matrix op
  return __builtin_amdgcn_wmma_f32_16x16x4_f32(
      /*neg_a=*/false, a, /*neg_b=*/false, b,
      /*c_mod=*/(short)0, c, /*reuse_a=*/false, /*reuse_b=*/false);
}

// One workgroup per token t. 256 threads = 8 waves (wave32).
// Wave w owns output columns [16w, 16w+16) and both 16-row M tiles.
__global__ __launch_bounds__(256) void fused_token_kernel(
    const float* __restrict__ x,   // [B,C,D]
    const float* __restrict__ Wg,  // [T,D,O]
    const float* __restrict__ bg,  // [T,O]
    const float* __restrict__ Wm,  // [T,C]
    const float* __restrict__ Wp1, // [T,O,O]  (row = out, col = in)
    const float* __restrict__ bp1, // [T,O]
    const float* __restrict__ Wp2, // [T,O,O]
    const float* __restrict__ bp2, // [T,O]
    float* __restrict__ tokens_out, // [B,T,O]
    float* __restrict__ pt_out)     // [B,T,O]
{
  __shared__ float lds_xm[B_ * XM_STRIDE];   // ~66 KB
  __shared__ float lds_tok[B_ * TO_STRIDE];  // ~16.9 KB
  __shared__ float lds_h[B_ * TO_STRIDE];    // ~16.9 KB

  const int t    = blockIdx.x;
  const int tid  = threadIdx.x;
  const int lane = tid & 31;
  const int w    = tid >> 5;       // wave id 0..7
  const int half = lane >> 4;      // 0: lanes 0-15, 1: lanes 16-31
  const int l15  = lane & 15;
  const int ncol = w * 16 + l15;   // output column owned by this lane (0..127)

  // --- Wm row for this token (uniform -> scalar loads) ---
  float wm[C_];
  float sWm = 0.f;
#pragma unroll
  for (int c = 0; c < C_; ++c) { wm[c] = Wm[t * C_ + c]; sWm += wm[c]; }

  // --- Stage 0: xm[b][d] = sum_c wm[c] * x[b][c][d]  (contract C first) ---
  {
    const int b  = tid >> 3;            // 0..31
    const int d0 = (tid & 7) * 64;      // 64-float chunk per thread
    const float* xb = x + (size_t)b * C_ * D_ + d0;
#pragma unroll 4
    for (int i = 0; i < 16; ++i) {
      v4f acc = {};
#pragma unroll
      for (int c = 0; c < C_; ++c) {
        v4f xv = *(const v4f*)(xb + c * D_ + i * 4);
        acc += wm[c] * xv;
      }
      *(v4f*)&lds_xm[b * XM_STRIDE + d0 + i * 4] = acc;
    }
  }
  __syncthreads();

  // --- Stage 1: tokens = relu(xm[32,512] @ Wg[t][512,128] + sWm*bg[t]) ---
  {
    const float* Bg  = Wg + (size_t)t * D_ * O_;
    const float* a0p = &lds_xm[l15 * XM_STRIDE + half * 2];
    const float* a1p = &lds_xm[(16 + l15) * XM_STRIDE + half * 2];
    const float* bp  = Bg + (size_t)(half * 2) * O_ + ncol;
    v8f acc0 = {}, acc1 = {};
#pragma unroll 4
    for (int k = 0; k < D_; k += 4) {
      v2f a0 = *(const v2f*)(a0p + k);
      v2f a1 = *(const v2f*)(a1p + k);
      v2f b;
      b.x = __builtin_nontemporal_load(bp + (size_t)k * O_);        // row k+2*half
      b.y = __builtin_nontemporal_load(bp + (size_t)(k + 1) * O_);  // row k+2*half+1
      acc0 = wmma_f32_16x16x4(a0, b, acc0);
      acc1 = wmma_f32_16x16x4(a1, b, acc1);
    }
    const float biasv = sWm * bg[t * O_ + ncol];
#pragma unroll
    for (int r = 0; r < 8; ++r) {
      const int b0 = r + half * 8;   // row within 16-row tile
      const float v0 = fmaxf(acc0[r] + biasv, 0.f);
      const float v1 = fmaxf(acc1[r] + biasv, 0.f);
      lds_tok[b0 * TO_STRIDE + ncol]        = v0;
      lds_tok[(16 + b0) * TO_STRIDE + ncol] = v1;
      tokens_out[(size_t)b0 * T_ * O_ + (size_t)t * O_ + ncol]        = v0;
      tokens_out[(size_t)(16 + b0) * T_ * O_ + (size_t)t * O_ + ncol] = v1;
    }
  }
  __syncthreads();

  // --- Stage 2: h = relu(tokens @ Wp1[t]^T + bp1[t]) ; B[k][n] = Wp1[n*O+k] ---
  {
    const float* W1  = Wp1 + (size_t)t * O_ * O_;
    const float* a0p = &lds_tok[l15 * TO_STRIDE + half * 2];
    const float* a1p = &lds_tok[(16 + l15) * TO_STRIDE + half * 2];
    const float* bpp = W1 + (size_t)ncol * O_ + half * 2;
    v8f acc0 = {}, acc1 = {};
#pragma unroll 4
    for (int k = 0; k < O_; k += 4) {
      v2f a0 = *(const v2f*)(a0p + k);
      v2f a1 = *(const v2f*)(a1p + k);
      v2f b  = __builtin_nontemporal_load((const v2f*)(bpp + k));
      acc0 = wmma_f32_16x16x4(a0, b, acc0);
      acc1 = wmma_f32_16x16x4(a1, b, acc1);
    }
    const float biasv = bp1[t * O_ + ncol];
#pragma unroll
    for (int r = 0; r < 8; ++r) {
      const int b0 = r + half * 8;
      lds_h[b0 * TO_STRIDE + ncol]        = fmaxf(acc0[r] + biasv, 0.f);
      lds_h[(16 + b0) * TO_STRIDE + ncol] = fmaxf(acc1[r] + biasv, 0.f);
    }
  }
  __syncthreads();

  // --- Stage 3: pt = h @ Wp2[t]^T + bp2[t] (no relu) -> d_out pt section ---
  {
    const float* W2  = Wp2 + (size_t)t * O_ * O_;
    const float* a0p = &lds_h[l15 * TO_STRIDE + half * 2];
    const float* a1p = &lds_h[(16 + l15) * TO_STRIDE + half * 2];
    const float* bpp = W2 + (size_t)ncol * O_ + half * 2;
    v8f acc0 = {}, acc1 = {};
#pragma unroll 4
    for (int k = 0; k < O_; k += 4) {
      v2f a0 = *(const v2f*)(a0p + k);
      v2f a1 = *(const v2f*)(a1p + k);
      v2f b  = __builtin_nontemporal_load((const v2f*)(bpp + k));
      acc0 = wmma_f32_16x16x4(a0, b, acc0);
      acc1 = wmma_f32_16x16x4(a1, b, acc1);
    }
    const float biasv = bp2[t * O_ + ncol];
#pragma unroll
    for (int r = 0; r < 8; ++r) {
      const int b0 = r + half * 8;
      pt_out[(size_t)b0 * T_ * O_ + (size_t)t * O_ + ncol]        = acc0[r] + biasv;
      pt_out[(size_t)(16 + b0) * T_ * O_ + (size_t)t * O_ + ncol] = acc1[r] + biasv;
    }
  }
}

// out[b,e,o] = sum_t Wa[e,t] * pt[b,t,o]   (pt is 4.2 MB -> L2-resident)
__global__ __launch_bounds__(256) void aggregate_kernel(
    const float* __restrict__ Wa,   // [E,T]
    const float* __restrict__ pt,   // [B,T,O]
    float* __restrict__ out)        // [B,E,O]
{
  const int g = blockIdx.x * blockDim.x + threadIdx.x;  // 0 .. B*E*O-1
  const int o = g & (O_ - 1);
  const int e = (g >> 7) & (E_ - 1);
  const int b = g >> 11;
  const float* p  = pt + (size_t)b * T_ * O_ + o;
  const float* wa = Wa + e * T_;
  float acc = 0.f;
#pragma unroll 8
  for (int t = 0; t < T_; ++t) acc += wa[t] * p[(size_t)t * O_];
  out[g] = acc;  // g == b*E*O + e*O + o
}

extern "C" void kernel_launch(void* const* d_in, const int* in_sizes, int n_in,
                              void* d_out, int out_size, void* d_ws, size_t ws_size,
                              hipStream_t stream) {
  (void)in_sizes; (void)n_in; (void)out_size; (void)d_ws; (void)ws_size;
  const float* x   = (const float*)d_in[0];
  const float* Wg  = (const float*)d_in[1];
  const float* bg  = (const float*)d_in[2];
  const float* Wm  = (const float*)d_in[3];
  const float* Wp1 = (const float*)d_in[4];
  const float* bp1 = (const float*)d_in[5];
  const float* Wp2 = (const float*)d_in[6];
  const float* bp2 = (const float*)d_in[7];
  const float* Wa  = (const float*)d_in[8];

  float* out    = (float*)d_out;            // [B,E,O]
  float* tokens = out + B_ * E_ * O_;       // [B,T,O]
  float* pt     = tokens + B_ * T_ * O_;    // [B,T,O]

  fused_token_kernel<<<T_, 256, 0, stream>>>(x, Wg, bg, Wm, Wp1, bp1, Wp2, bp2,
                                             tokens, pt);
  aggregate_kernel<<<(B_ * E_ * O_) / 256, 256, 0, stream>>>(Wa, pt, out);
}